// FirstShared_88467736363911
// MI455X (gfx1250) — compile-verified
//
#include <hip/hip_runtime.h>

// ---------------------------------------------------------------------------
// GCN 2-layer inference for MI455X (gfx1250, wave32).
//   h0 = x @ w1                      (WMMA f32 16x16x4)
//   h1 = scatter_add(edge_val * h0[col]) by row   (fp32 atomics, L2-resident)
//   h2 = relu(h1) @ w2               (WMMA, relu fused into A-load)
//   out = scatter_add(edge_val * h2[col]) by row
// ---------------------------------------------------------------------------

typedef float v2f __attribute__((ext_vector_type(2)));
typedef float v8f __attribute__((ext_vector_type(8)));

#ifndef GCN_DFEAT
#define GCN_DFEAT 128
#define GCN_C1    64
#define GCN_C2    32
#endif

// C[M x NW] = op(A)[M x K] @ W[K x NW], one wave per 16x16 tile.
// blockDim.x = 32 * (NW/16); gridDim.x = M/16 (M must be a multiple of 16).
// WMMA f32 16x16x4 fragment layouts (ISA 7.12.2):
//   A (16x4): lane&15 = M row; lane>=16 holds K+2; v2f = {K, K+1}
//   B (4x16): lane&15 = N col; lane>=16 holds K+2; v2f = {row K, row K+1}
//   C/D (16x16): vgpr r, lane l -> M = r + 8*(l>>4), N = l&15
template <int K, int NW, bool RELU_A>
__global__ __launch_bounds__(32 * (NW / 16)) void gemm_wmma_f32(
    const float* __restrict__ A, const float* __restrict__ W,
    float* __restrict__ C)
{
    const int lane = threadIdx.x & 31;
    const int wave = threadIdx.x >> 5;          // which 16-col tile
    const int m    = lane & 15;
    const int hi   = lane >> 4;                 // 0: K+{0,1}, 1: K+{2,3}
    const long aRow = (long)(blockIdx.x * 16 + m) * K;
    const int  nCol = wave * 16 + m;            // column index for B and C

    v8f acc = {};

#pragma unroll 4
    for (int k0 = 0; k0 < K; k0 += 4) {
        const int ka = k0 + (hi << 1);

        float a0 = A[aRow + ka + 0];
        float a1 = A[aRow + ka + 1];
        if (RELU_A) {
            a0 = fmaxf(a0, 0.0f);
            a1 = fmaxf(a1, 0.0f);
        }
        v2f av; av.x = a0; av.y = a1;

        v2f bv;
        bv.x = W[(long)(ka + 0) * NW + nCol];
        bv.y = W[(long)(ka + 1) * NW + nCol];

        // 8 args: (neg_a, A, neg_b, B, c_mod, C, reuse_a, reuse_b)
        acc = __builtin_amdgcn_wmma_f32_16x16x4_f32(
            false, av, false, bv, (short)0, acc, false, false);
    }

#pragma unroll
    for (int r = 0; r < 8; ++r) {
        const int mOut = r + (hi << 3);
        C[(long)(blockIdx.x * 16 + mOut) * NW + nCol] = acc[r];
    }
}

// out[row[e], :] += val[e] * h[col[e], :]
// F = 4 << LOGF4 features per row; each lane handles one float4 chunk, so
// (1 << LOGF4) consecutive lanes share one edge -> coalesced 128B+ gathers.
template <int LOGF4>
__global__ void spmm_atomic(const int* __restrict__ erow,
                            const int* __restrict__ ecol,
                            const float* __restrict__ eval,
                            const float* __restrict__ h,
                            float* __restrict__ out, int E)
{
    const int  F4    = 1 << LOGF4;
    const int  F     = F4 * 4;
    const long total = (long)E << LOGF4;
    const long stride = (long)gridDim.x * blockDim.x;

    for (long idx = (long)blockIdx.x * blockDim.x + threadIdx.x;
         idx < total; idx += stride) {
        const int e = (int)(idx >> LOGF4);
        const int f = ((int)idx & (F4 - 1)) * 4;

        const int   c = ecol[e];
        const int   r = erow[e];
        const float v = eval[e];

        const float4 hv = *(const float4*)(h + (long)c * F + f);
        float* o = out + (long)r * F + f;
        atomicAdd(o + 0, v * hv.x);
        atomicAdd(o + 1, v * hv.y);
        atomicAdd(o + 2, v * hv.z);
        atomicAdd(o + 3, v * hv.w);
    }
}

extern "C" void kernel_launch(void* const* d_in, const int* in_sizes, int n_in,
                              void* d_out, int out_size, void* d_ws, size_t ws_size,
                              hipStream_t stream)
{
    const float* x    = (const float*)d_in[0];   // [N, 128]
    const int*   erow = (const int*)  d_in[1];   // [E]
    const int*   ecol = (const int*)  d_in[2];   // [E]
    const float* eval = (const float*)d_in[3];   // [E]
    const float* w1   = (const float*)d_in[4];   // [128, 64]
    const float* w2   = (const float*)d_in[5];   // [64, 32]
    float*       out  = (float*)d_out;           // [N, 32]

    const int N = in_sizes[0] / GCN_DFEAT;       // 100000 (multiple of 16)
    const int E = in_sizes[1];                   // 1600000

    float* h0 = (float*)d_ws;                    // [N, 64]  x @ w1
    float* h1 = h0 + (size_t)N * GCN_C1;         // [N, 64]  spmm1 result
    float* h2 = h0;                              // [N, 32]  relu(h1) @ w2 (reuse)

    // ---- layer 1 ----
    hipMemsetAsync(h1, 0, (size_t)N * GCN_C1 * sizeof(float), stream);

    gemm_wmma_f32<GCN_DFEAT, GCN_C1, false>
        <<<N / 16, 32 * (GCN_C1 / 16), 0, stream>>>(x, w1, h0);

    {
        const long total  = (long)E << 4;                     // E * 16 chunks
        const int  blocks = (int)((total + 255) / 256);
        spmm_atomic<4><<<blocks, 256, 0, stream>>>(erow, ecol, eval, h0, h1, E);
    }

    // ---- layer 2 (relu fused into GEMM A-load) ----
    gemm_wmma_f32<GCN_C1, GCN_C2, true>
        <<<N / 16, 32 * (GCN_C2 / 16), 0, stream>>>(h1, w2, h2);

    hipMemsetAsync(out, 0, (size_t)out_size * sizeof(float), stream);

    {
        const long total  = (long)E << 3;                     // E * 8 chunks
        const int  blocks = (int)((total + 255) / 256);
        spmm_atomic<3><<<blocks, 256, 0, stream>>>(erow, ecol, eval, h2, out, E);
    }
}